// adder2d_26568667693193
// MI455X (gfx1250) — compile-verified
//
#include <hip/hip_runtime.h>
#include <hip/hip_bf16.h>

// AdderNet conv2d, 3x3, stride 1, pad 1.
// x: [32,64,32,32] f32, W: [64,64,3,3] f32, out: [32,64,32,32] f32
// out[n,f,h,w] = -sum_{c,i,j} |W[f,c,i,j] - x_pad[n,c,h+i-1,w+j-1]|

#define KK        3
#define C_IN      64
#define F_OUT     64
#define HW        32
#define N_IMG     32
#define TH        8            // output rows per block
#define FPB       16           // filters per block
#define XROW      40           // padded LDS row stride in floats (interior 16B aligned)
#define XS_FLOATS (C_IN * 10 * XROW)   // 25600 floats = 100 KB
#define WS_FLOATS (FPB * C_IN * 12)    // 12288 floats = 48 KB
#define LDS_BYTES ((XS_FLOATS + WS_FLOATS) * 4)

__device__ __forceinline__ unsigned lds_off_u32(const void* p) {
  // Generic LDS addresses carry the LDS byte address in addr[31:0] (ISA 10.2).
  return (unsigned)(unsigned long long)p;
}

__global__ __launch_bounds__(256)
void adder2d_26568667693193_kernel(const float* __restrict__ x,
                                   const float* __restrict__ w,
                                   float* __restrict__ out) {
  extern __shared__ float smem[];
  float* xs = smem;               // [C_IN][10][XROW]  padded input tile
  float* ws = smem + XS_FLOATS;   // [FPB][C_IN][12]   weights, 48B row stride

  const int tid  = threadIdx.x;
  const int row0 = blockIdx.x * TH;
  const int f0   = blockIdx.y * FPB;
  const int n    = blockIdx.z;

  // ---- phase 1: zero the padded input tile (provides zero padding) ----
  {
    float4 z4 = make_float4(0.f, 0.f, 0.f, 0.f);
    float4* xs4 = (float4*)xs;
    #pragma unroll
    for (int i = 0; i < XS_FLOATS / 4 / 256; ++i)   // 25 iters, exact
      xs4[tid + i * 256] = z4;
  }
  __syncthreads();

  // ---- phase 2a: async copy weights for this filter group (contiguous in W) ----
  {
    const float* wsrc = w + (size_t)f0 * (C_IN * 9);
    for (int e = tid; e < FPB * C_IN * 9; e += 256) {   // 36 iters, exact
      int row = e / 9;
      int t   = e - row * 9;
      unsigned dst   = lds_off_u32(ws + row * 12 + t);
      const float* g = wsrc + e;
      asm volatile("global_load_async_to_lds_b32 %0, %1, off"
                   :: "v"(dst), "v"(g) : "memory");
    }
  }

  // ---- phase 2b: async copy the input tile interior (16B-aligned b128) ----
  {
    const float* xsrc = x + (size_t)n * C_IN * HW * HW;
    for (int dy = 0; dy < 10; ++dy) {
      int r = row0 - 1 + dy;
      if (0 <= r && r < HW) {
        for (int e = tid; e < C_IN * 8; e += 256) {     // 2 iters, exact
          int c = e >> 3, q = e & 7;
          unsigned dst   = lds_off_u32(xs + (c * 10 + dy) * XROW + 4 + q * 4);
          const float* g = xsrc + (size_t)c * (HW * HW) + r * HW + q * 4;
          asm volatile("global_load_async_to_lds_b128 %0, %1, off"
                       :: "v"(dst), "v"(g) : "memory");
        }
      }
    }
  }

  asm volatile("s_wait_asynccnt 0" ::: "memory");
  __syncthreads();

  // ---- phase 3: compute. One output pixel x FPB filters per thread. ----
  const int px = tid & 31;   // output column
  const int py = tid >> 5;   // output row within tile

  float acc[FPB];
  #pragma unroll
  for (int fl = 0; fl < FPB; ++fl) acc[fl] = 0.f;

  for (int c = 0; c < C_IN; ++c) {
    // 3x3 window for this channel: lds col = 4 + (px + dx - 1), rows py..py+2
    float xv[9];
    #pragma unroll
    for (int dy = 0; dy < 3; ++dy)
      #pragma unroll
      for (int dx = 0; dx < 3; ++dx)
        xv[dy * 3 + dx] = xs[(c * 10 + py + dy) * XROW + 3 + px + dx];

    #pragma unroll
    for (int fl = 0; fl < FPB; ++fl) {
      const float* wr = ws + (fl * C_IN + c) * 12;     // 48B aligned
      float4 wa = *(const float4*)wr;                  // ds_load_b128
      float4 wb = *(const float4*)(wr + 4);            // ds_load_b128
      float  w8 = wr[8];                               // ds_load_b32 (broadcast)
      float s = fabsf(wa.x - xv[0]) + fabsf(wa.y - xv[1]) + fabsf(wa.z - xv[2])
              + fabsf(wa.w - xv[3]) + fabsf(wb.x - xv[4]) + fabsf(wb.y - xv[5])
              + fabsf(wb.z - xv[6]) + fabsf(wb.w - xv[7]) + fabsf(w8  - xv[8]);
      acc[fl] += s;
    }
  }

  // ---- write out: out[n][f0+fl][row0+py][px] = -acc ----
  float* op = out + (((size_t)n * F_OUT + f0) * HW + (row0 + py)) * HW + px;
  #pragma unroll
  for (int fl = 0; fl < FPB; ++fl)
    op[(size_t)fl * (HW * HW)] = -acc[fl];
}

extern "C" void kernel_launch(void* const* d_in, const int* in_sizes, int n_in,
                              void* d_out, int out_size, void* d_ws, size_t ws_size,
                              hipStream_t stream) {
  (void)in_sizes; (void)n_in; (void)out_size; (void)d_ws; (void)ws_size;
  const float* x = (const float*)d_in[0];
  const float* w = (const float*)d_in[1];
  float* out     = (float*)d_out;

  dim3 grid(HW / TH, F_OUT / FPB, N_IMG);   // 4 x 4 x 32 = 512 blocks
  adder2d_26568667693193_kernel<<<grid, 256, LDS_BYTES, stream>>>(x, w, out);
}